// DeformableDETR_11484742549692
// MI455X (gfx1250) — compile-verified
//
#include <hip/hip_runtime.h>
#include <math.h>

// ---------------- problem constants (from reference) ----------------
#define NHEADS  8
#define NLEV    4
#define NPTS    4
#define DMODEL  256
#define HDIM    32
#define LEN_IN  21760
#define BATCH   2
#define MROWS   (BATCH * LEN_IN)   // 43520 query rows
#define GEMM_K  256                // K is 256 in all three GEMMs

typedef float v2f __attribute__((ext_vector_type(2)));
typedef float v8f __attribute__((ext_vector_type(8)));

// =====================================================================
// Row-major GEMM  C[M,N] = A[M,K] * W[K,N] + bias[N]   (K == 256)
// V_WMMA_F32_16X16X4_F32, exact fp32 (pipeline is bandwidth bound).
//
// Block = 128 threads = 4 waves. Block tile = 128 rows x 64 cols.
// Each wave: 32 rows x 64 cols = 8 v8f accumulators.
//
// B strip W[0:256][n0:n0+64] is staged ONCE into LDS (64 KB),
// pre-swizzled into WMMA fragment order so the inner loop reads one
// lane-linear ds_load_b64 per fragment (bank-conflict free: 32 lanes
// x 8B = one full sweep of the 64 LDS banks).
//
// f32 WMMA fragment layouts (ISA 7.12.2):
//   A 16x4 : lane&15 = M row ; 2 VGPRs = K { (lane>>4)*2, +1 }
//   B 4x16 : lane&15 = N col ; 2 VGPRs = K { (lane>>4)*2, +1 }
//   C 16x16: VGPR v -> M = v + (lane>>4)*8 ; N = lane&15
// =====================================================================
__global__ __launch_bounds__(128) void gemm_wmma_f32(
    const float* __restrict__ A, const float* __restrict__ W,
    const float* __restrict__ bias, float* __restrict__ C,
    int M, int N)
{
    // [kstep 0..63][tile 0..3][lane 0..31] -> v2f fragment element
    __shared__ v2f bsh[64 * 4 * 32];          // 64 KB

    const int wave = threadIdx.x >> 5;
    const int lane = threadIdx.x & 31;
    const int l15  = lane & 15;
    const int lhi  = lane >> 4;               // 0|1 -> K pair select
    const int m0   = blockIdx.x * 128 + wave * 32;
    const int n0   = blockIdx.y * 64;

    // ---- stage B strip into LDS in fragment order (64 frags / wave) ----
    {
        const int fbase = wave * 64;
#pragma unroll 4
        for (int i = 0; i < 64; ++i) {
            const int f  = fbase + i;
            const int kk = f >> 2;            // k-step
            const int t  = f & 3;             // 16-col tile
            const int kr = (kk << 2) + (lhi << 1);
            const float* wp = W + (size_t)kr * N + n0 + t * 16 + l15;
            v2f b;
            b.x = wp[0];
            b.y = wp[N];
            bsh[f * 32 + lane] = b;           // ds_store_b64, lane-linear
        }
    }
    __syncthreads();

    v8f acc[8] = {};

    const float* arow0 = A + (size_t)(m0 + l15) * GEMM_K + (lhi << 1);
    const float* arow1 = arow0 + (size_t)16 * GEMM_K;

    // software pipeline: prefetch A fragments for the next k-step
    v2f a0, a1, a0n, a1n;
    a0 = *(const v2f*)(arow0);
    a1 = *(const v2f*)(arow1);

    for (int kk = 0; kk < 64; ++kk) {
        if (kk < 63) {                         // issue next A loads early
            a0n = *(const v2f*)(arow0 + (kk + 1) * 4);
            a1n = *(const v2f*)(arow1 + (kk + 1) * 4);
        }
        const v2f* bfr = &bsh[(kk << 2) * 32 + lane];
#pragma unroll
        for (int t = 0; t < 4; ++t) {
            const v2f b = bfr[t * 32];         // ds_load_b64, conflict-free
            acc[t]     = __builtin_amdgcn_wmma_f32_16x16x4_f32(
                false, a0, false, b, (short)0, acc[t],     false, false);
            acc[4 + t] = __builtin_amdgcn_wmma_f32_16x16x4_f32(
                false, a1, false, b, (short)0, acc[4 + t], false, false);
        }
        a0 = a0n;
        a1 = a1n;
    }

    // ---- epilogue: bias + store ----
#pragma unroll
    for (int t = 0; t < 4; ++t) {
        const int n  = n0 + t * 16 + l15;
        const float bv = bias[n];
#pragma unroll
        for (int r = 0; r < 2; ++r) {
            const int mb = m0 + r * 16 + (lhi << 3);
#pragma unroll
            for (int v = 0; v < 8; ++v) {
                C[(size_t)(mb + v) * N + n] = acc[r * 4 + t][v] + bv;
            }
        }
    }
}

// =====================================================================
// Deformable sampling: one wave32 per (b, q, head); lane = channel d.
// Softmax over the 16 (level,point) logits is computed per-lane on
// wave-uniform data (one cacheline). Each bilinear corner read is a
// coalesced 32x4B = 128B line: feat[b, row, h*32 + lane].
// Feature map (44.6 MB) is resident in the 192 MB L2.
// =====================================================================
__global__ __launch_bounds__(256) void msda_sample(
    const float* __restrict__ refp,     // [B, LEN_IN, 2]
    const float* __restrict__ feat,     // [B, LEN_IN, 256]
    const float* __restrict__ off,      // [MROWS, 256] = h*32 + s*2 + {x,y}
    const float* __restrict__ logits,   // [MROWS, 128] = h*16 + s
    const int*   __restrict__ spatial,  // [4][2] (H, W)
    const int*   __restrict__ lstart,   // [4]
    float*       __restrict__ outh)     // [MROWS, 256] = h*32 + d
{
    const int wid  = blockIdx.x * 8 + (threadIdx.x >> 5);
    const int lane = threadIdx.x & 31;
    if (wid >= MROWS * NHEADS) return;

    const int h  = wid & (NHEADS - 1);
    const int bq = wid >> 3;
    const int b  = bq / LEN_IN;

    const float rx = refp[(size_t)bq * 2 + 0];
    const float ry = refp[(size_t)bq * 2 + 1];
    const float* orow = off    + (size_t)bq * 256 + h * 32;
    const float* lrow = logits + (size_t)bq * 128 + h * 16;

    // ---- softmax over 16 logits (uniform across the wave) ----
    float l[16];
    float mx = -3.0e38f;
#pragma unroll
    for (int i = 0; i < 16; ++i) { l[i] = lrow[i]; mx = fmaxf(mx, l[i]); }
    float ssum = 0.f;
#pragma unroll
    for (int i = 0; i < 16; ++i) { l[i] = __expf(l[i] - mx); ssum += l[i]; }
    const float inv = 1.f / ssum;

    float acc = 0.f;

#pragma unroll
    for (int lvl = 0; lvl < NLEV; ++lvl) {
        const int Hh = spatial[lvl * 2 + 0];
        const int Ww = spatial[lvl * 2 + 1];
        const int st = lstart[lvl];
        const float fW = (float)Ww, fH = (float)Hh;
        const float* fbase =
            feat + ((size_t)b * LEN_IN + st) * DMODEL + h * HDIM + lane;

#pragma unroll
        for (int p = 0; p < NPTS; ++p) {
            const int s = lvl * NPTS + p;
            // loc = ref + off/(W,H); pixel = loc*size - 0.5  =>  rx*W + ox - 0.5
            const float x = rx * fW + orow[2 * s + 0] - 0.5f;
            const float y = ry * fH + orow[2 * s + 1] - 0.5f;
            const float x0f = floorf(x), y0f = floorf(y);
            const float fx = x - x0f, fy = y - y0f;
            const int x0 = (int)x0f, y0 = (int)y0f;

            const float aw  = l[s] * inv;
            const float w00 = (1.f - fx) * (1.f - fy) * aw;
            const float w01 = fx * (1.f - fy) * aw;
            const float w10 = (1.f - fx) * fy * aw;
            const float w11 = fx * fy * aw;

            const bool vx0 = ((unsigned)x0       < (unsigned)Ww);
            const bool vx1 = ((unsigned)(x0 + 1) < (unsigned)Ww);
            const bool vy0 = ((unsigned)y0       < (unsigned)Hh);
            const bool vy1 = ((unsigned)(y0 + 1) < (unsigned)Hh);

            if (vx0 & vy0) acc += w00 * fbase[(size_t)(y0 * Ww + x0)           * DMODEL];
            if (vx1 & vy0) acc += w01 * fbase[(size_t)(y0 * Ww + x0 + 1)       * DMODEL];
            if (vx0 & vy1) acc += w10 * fbase[(size_t)((y0 + 1) * Ww + x0)     * DMODEL];
            if (vx1 & vy1) acc += w11 * fbase[(size_t)((y0 + 1) * Ww + x0 + 1) * DMODEL];
        }
    }

    outh[(size_t)bq * 256 + h * HDIM + lane] = acc;
}

// =====================================================================
extern "C" void kernel_launch(void* const* d_in, const int* in_sizes, int n_in,
                              void* d_out, int out_size, void* d_ws, size_t ws_size,
                              hipStream_t stream)
{
    const float* query  = (const float*)d_in[0];
    const float* refp   = (const float*)d_in[1];
    const float* feat   = (const float*)d_in[2];
    const int*   spat   = (const int*)  d_in[3];
    const int*   lstart = (const int*)  d_in[4];
    const float* Woff   = (const float*)d_in[5];
    const float* boff   = (const float*)d_in[6];
    const float* Wattn  = (const float*)d_in[7];
    const float* battn  = (const float*)d_in[8];
    const float* Wout   = (const float*)d_in[9];
    const float* bout   = (const float*)d_in[10];

    // workspace: off (44.6MB) | logits (22.3MB) | per-head out (44.6MB)
    float* off   = (float*)d_ws;
    float* logit = off   + (size_t)MROWS * 256;
    float* heads = logit + (size_t)MROWS * 128;

    const dim3 blk(128);
    // 1) sampling-offset projection: [43520,256] x [256,256]
    gemm_wmma_f32<<<dim3(MROWS / 128, 256 / 64), blk, 0, stream>>>(
        query, Woff, boff, off, MROWS, 256);
    // 2) attention-logit projection: [43520,256] x [256,128]
    gemm_wmma_f32<<<dim3(MROWS / 128, 128 / 64), blk, 0, stream>>>(
        query, Wattn, battn, logit, MROWS, 128);
    // 3) softmax + bilinear deformable sampling (wave per (b,q,head))
    msda_sample<<<(MROWS * NHEADS) / 8, 256, 0, stream>>>(
        refp, feat, off, logit, spat, lstart, heads);
    // 4) output projection: [43520,256] x [256,256]
    gemm_wmma_f32<<<dim3(MROWS / 128, 256 / 64), blk, 0, stream>>>(
        heads, Wout, bout, (float*)d_out, MROWS, 256);
}